// GCN_17343077941655
// MI455X (gfx1250) — compile-verified
//
#include <hip/hip_runtime.h>

// ---------------------------------------------------------------------------
// GCN (2-layer, PyG-style) for MI455X / gfx1250, wave32.
//   layer: g[u] = dinv[u] * (X @ W)[u]            (WMMA f32 16x16x4 GEMM)
//          acc[v] = g[v] + sum_{u->v} g[u]        (self-loop init + atomics)
//          out[v] = act(dinv[v]*acc[v] + b)
// ---------------------------------------------------------------------------

typedef __attribute__((ext_vector_type(2))) float v2f;
typedef __attribute__((ext_vector_type(8))) float v8f;

#define FEAT_IN 128
#define F1 64
#define F2 32

// ---------------- degree / normalization ----------------------------------

__global__ void k_init_deg(float* __restrict__ deg, int n) {
  int i = blockIdx.x * blockDim.x + threadIdx.x;
  if (i < n) deg[i] = 1.0f;  // self loop contributes 1 to every node's degree
}

__global__ void k_count_deg(const int* __restrict__ dst, float* __restrict__ deg, int E) {
  int e = blockIdx.x * blockDim.x + threadIdx.x;
  if (e < E) atomicAdd(&deg[dst[e]], 1.0f);
}

__global__ void k_rsqrt(float* __restrict__ deg, int n) {
  int i = blockIdx.x * blockDim.x + threadIdx.x;
  if (i < n) deg[i] = rsqrtf(deg[i]);  // deg >= 1 always (self loops)
}

// ---------------- WMMA GEMM with per-row scale epilogue --------------------
// G[row, :] = dinv[row] * (X[row, :] @ W)
// Block = 256 threads = 8 waves; wave w handles rows [blk*128 + w*16, +16),
// all FOUT columns (FOUT/16 n-tiles), K accumulated 4 at a time with
// V_WMMA_F32_16X16X4_F32.
//
// W is staged in LDS in a *fragment-packed* layout: for each
// (kstep = k0/4, ksel, col) the pair (W[k][col], W[k+1][col]) with
// k = k0 + 2*ksel is stored contiguously, so each lane's B fragment is a
// single aligned ds_load_b64 straight into an even VGPR pair (no repack
// moves).  Row stride S = FOUT+16 float2 makes the ksel=1 half-wave land on
// banks 32..63 while ksel=0 uses banks 0..31 -> conflict-free b64 access.

template <int K, int FOUT>
__global__ void __launch_bounds__(256)
k_gemm_scale(const float* __restrict__ X, const float* __restrict__ W,
             const float* __restrict__ dinv, float* __restrict__ G, int n) {
  constexpr int S = FOUT + 16;      // float2 stride per (kstep,ksel) row
  constexpr int NROWS = K / 2;      // (K/4 ksteps) * 2 ksel
  __shared__ float sW[NROWS * S * 2];

  for (int i = threadIdx.x; i < K * FOUT; i += 256) {
    const int k = i / FOUT;
    const int c = i - k * FOUT;
    const int kstep = k >> 2;
    const int kse   = (k >> 1) & 1;
    const int half  = k & 1;
    sW[((kstep * 2 + kse) * S + c) * 2 + half] = W[i];
  }
  __syncthreads();

  const int lane = threadIdx.x & 31;
  const int wave = threadIdx.x >> 5;
  const int m    = lane & 15;   // row within 16-row tile / col within n-tile
  const int ksel = lane >> 4;   // which K-pair this lane holds (A/B layout)

  const int wave_row0 = blockIdx.x * 128 + wave * 16;
  int arow = wave_row0 + m;
  if (arow > n - 1) arow = n - 1;  // clamp: EXEC must stay all-1s for WMMA
  const float* xrow = X + (long)arow * K;

  // per-lane LDS base: ksel row select + col (= m) within row; +t*16 cols,
  // +kstep*2*S rows added in the loop.
  const float* sWlane = sW + (ksel * S + m) * 2;

  v8f acc[FOUT / 16] = {};

#pragma unroll
  for (int k0 = 0; k0 < K; k0 += 4) {
    const int kstep = k0 >> 2;
    // A fragment: lane l -> (M = l%16, K = k0 + 2*(l/16) + {0,1})
    const v2f a = *(const v2f*)(xrow + k0 + 2 * ksel);
#pragma unroll
    for (int t = 0; t < FOUT / 16; ++t) {
      // B fragment: one aligned 8-byte LDS load per wmma
      const v2f b = *(const v2f*)(sWlane + (kstep * 2 * S + t * 16) * 2);
      acc[t] = __builtin_amdgcn_wmma_f32_16x16x4_f32(
          /*neg_a=*/false, a, /*neg_b=*/false, b,
          /*c_mod=*/(short)0, acc[t], /*reuse_a=*/false, /*reuse_b=*/false);
    }
  }

  // D layout: VGPR r, lanes 0-15 -> M=r, lanes 16-31 -> M=8+r; N = lane%16.
#pragma unroll
  for (int r = 0; r < 8; ++r) {
    const int rr = wave_row0 + ksel * 8 + r;
    if (rr < n) {
      const float sc = dinv[rr];
#pragma unroll
      for (int t = 0; t < FOUT / 16; ++t)
        G[(long)rr * FOUT + t * 16 + m] = sc * acc[t][r];
    }
  }
}

// ---------------- aggregation ----------------------------------------------

__global__ void k_copy4(const float4* __restrict__ a, float4* __restrict__ b, long n4) {
  long i = (long)blockIdx.x * blockDim.x + threadIdx.x;
  if (i < n4) b[i] = a[i];
}

template <int FOUT>
__global__ void k_scatter(const int* __restrict__ src, const int* __restrict__ dst,
                          const float* __restrict__ g, float* __restrict__ acc, int E) {
  const int TPE = FOUT / 4;  // threads per edge, each moves a float4
  long t = (long)blockIdx.x * blockDim.x + threadIdx.x;
  long e = t / TPE;
  if (e >= E) return;
  const int f = (int)(t - e * TPE) * 4;
  const int s = src[e];
  const int d = dst[e];
  const float4 v = *(const float4*)(g + (long)s * FOUT + f);
  float* o = acc + (long)d * FOUT + f;
  atomicAdd(o + 0, v.x);
  atomicAdd(o + 1, v.y);
  atomicAdd(o + 2, v.z);
  atomicAdd(o + 3, v.w);
}

template <int FOUT, bool RELU>
__global__ void k_finalize(const float* __restrict__ acc, const float* __restrict__ dinv,
                           const float* __restrict__ bias, float* __restrict__ out, int n) {
  long i = (long)blockIdx.x * blockDim.x + threadIdx.x;
  if (i >= (long)n * FOUT) return;
  const int row = (int)(i / FOUT);  // FOUT is a power of two -> shift
  const int col = (int)(i & (FOUT - 1));
  float v = dinv[row] * acc[i] + bias[col];
  if (RELU) v = fmaxf(v, 0.0f);
  out[i] = v;
}

// ---------------- driver ----------------------------------------------------

extern "C" void kernel_launch(void* const* d_in, const int* in_sizes, int n_in,
                              void* d_out, int out_size, void* d_ws, size_t ws_size,
                              hipStream_t stream) {
  const float* x  = (const float*)d_in[0];
  const int*   ei = (const int*)d_in[1];
  const float* W1 = (const float*)d_in[2];
  const float* b1 = (const float*)d_in[3];
  const float* W2 = (const float*)d_in[4];
  const float* b2 = (const float*)d_in[5];
  float* out = (float*)d_out;

  const int N = in_sizes[0] / FEAT_IN;  // 50000
  const int E = in_sizes[1] / 2;        // 800000
  const int* srcI = ei;      // edge_index[0]
  const int* dstI = ei + E;  // edge_index[1]

  // workspace layout (floats):
  //   dinv : NPAD
  //   bufA : N*F1   (g1, later reused as h1)
  //   bufB : N*F1   (acc1, later reused as g2)
  //   bufC : N*F2   (acc2)
  float* ws = (float*)d_ws;
  const long NPAD = ((long)N + 63) & ~63L;
  float* dinv = ws;
  float* bufA = dinv + NPAD;
  float* bufB = bufA + (long)N * F1;
  float* bufC = bufB + (long)N * F1;

  const int BLK = 256;
  const int gN = (N + BLK - 1) / BLK;
  const int gE = (E + BLK - 1) / BLK;
  const int gGemm = (N + 127) / 128;

  // --- degree normalization ---
  k_init_deg<<<gN, BLK, 0, stream>>>(dinv, N);
  k_count_deg<<<gE, BLK, 0, stream>>>(dstI, dinv, E);
  k_rsqrt<<<gN, BLK, 0, stream>>>(dinv, N);

  // --- layer 1: g1 = dinv * (x @ W1) ; acc1 = g1 + scatter ; h1 = relu(dinv*acc1 + b1)
  const long n1 = (long)N * F1;
  k_gemm_scale<FEAT_IN, F1><<<gGemm, BLK, 0, stream>>>(x, W1, dinv, bufA, N);
  k_copy4<<<(int)((n1 / 4 + BLK - 1) / BLK), BLK, 0, stream>>>(
      (const float4*)bufA, (float4*)bufB, n1 / 4);
  {
    const long th = (long)E * (F1 / 4);
    k_scatter<F1><<<(int)((th + BLK - 1) / BLK), BLK, 0, stream>>>(srcI, dstI, bufA, bufB, E);
  }
  k_finalize<F1, true><<<(int)((n1 + BLK - 1) / BLK), BLK, 0, stream>>>(
      bufB, dinv, b1, bufA /* h1 overwrites g1 */, N);

  // --- layer 2: g2 = dinv * (h1 @ W2) ; acc2 = g2 + scatter ; out = dinv*acc2 + b2
  const long n2 = (long)N * F2;
  k_gemm_scale<F1, F2><<<gGemm, BLK, 0, stream>>>(bufA, W2, dinv, bufB /* g2 */, N);
  k_copy4<<<(int)((n2 / 4 + BLK - 1) / BLK), BLK, 0, stream>>>(
      (const float4*)bufB, (float4*)bufC, n2 / 4);
  {
    const long th = (long)E * (F2 / 4);
    k_scatter<F2><<<(int)((th + BLK - 1) / BLK), BLK, 0, stream>>>(srcI, dstI, bufB, bufC, E);
  }
  k_finalize<F2, false><<<(int)((n2 + BLK - 1) / BLK), BLK, 0, stream>>>(
      bufC, dinv, b2, out, N);
}